// CSRN_23888608100925
// MI455X (gfx1250) — compile-verified
//
#include <hip/hip_runtime.h>

// Problem constants (fixed by the reference: B=32, C=3, H=W=512)
#define NB   32
#define HH   512
#define WW   512
#define HWD  (HH * WW)          // 262144 = 1 << 18

typedef float v8f __attribute__((ext_vector_type(8)));
typedef float v2f __attribute__((ext_vector_type(2)));

__device__ __forceinline__ float fsig(float v) {
  // 1 / (1 + e^-v) with v_exp_f32 + v_rcp_f32
  return __builtin_amdgcn_rcpf(1.0f + __expf(-v));
}
__device__ __forceinline__ float ftanh_fast(float v) {
  // tanh(v) = 1 - 2/(e^{2v}+1); saturates correctly at +-inf
  return 1.0f - 2.0f * __builtin_amdgcn_rcpf(__expf(2.0f * v) + 1.0f);
}

// ---------------------------------------------------------------------------
// Directional scan: one 512-thread block per (batch, direction).
//   dir==0: scan over H (rows),   thread t = w position, conv couples over W
//   dir==1: scan over W (cols),   thread t = h position, conv couples over H
// Hidden state (3 floats) lives in registers; per-step GRU output goes to a
// double-buffered LDS line so the 3-tap conv needs one barrier per step.
// Next-step x is register-prefetched to hide global latency on the serial path.
// ---------------------------------------------------------------------------
__global__ __launch_bounds__(512) void csrn_scan(
    const float* __restrict__ x,
    const float* __restrict__ wih_r, const float* __restrict__ whh_r,
    const float* __restrict__ bih_r, const float* __restrict__ bhh_r,
    const float* __restrict__ cnw_r, const float* __restrict__ cnb_r,
    const float* __restrict__ wih_c, const float* __restrict__ whh_c,
    const float* __restrict__ bih_c, const float* __restrict__ bhh_c,
    const float* __restrict__ cnw_c, const float* __restrict__ cnb_c,
    float* __restrict__ ctx_above, float* __restrict__ ctx_left)
{
  const int b   = blockIdx.x;
  const int dir = blockIdx.y;        // uniform per block
  const int t   = threadIdx.x;       // 0..511 position along the scanned line

  const float* wih = dir ? wih_c : wih_r;
  const float* whh = dir ? whh_c : whh_r;
  const float* bih = dir ? bih_c : bih_r;
  const float* bhh = dir ? bhh_c : bhh_r;
  const float* cnw = dir ? cnw_c : cnw_r;
  const float* cnb = dir ? cnb_c : cnb_r;
  float* ctx = dir ? ctx_left : ctx_above;

  // Loop-invariant tiny weights -> registers (occupancy is irrelevant: 64 WGs)
  float Wi[9][3], Wh[9][3], Bi[9], Bh[9];
#pragma unroll
  for (int j = 0; j < 9; ++j) {
    Bi[j] = bih[j]; Bh[j] = bhh[j];
#pragma unroll
    for (int k = 0; k < 3; ++k) { Wi[j][k] = wih[j * 3 + k]; Wh[j][k] = whh[j * 3 + k]; }
  }
  float Cw[3][3][3], Cb[3];
#pragma unroll
  for (int o = 0; o < 3; ++o) {
    Cb[o] = cnb[o];
#pragma unroll
    for (int ci = 0; ci < 3; ++ci)
#pragma unroll
      for (int k = 0; k < 3; ++k) Cw[o][ci][k] = cnw[(o * 3 + ci) * 3 + k];
  }

  // Double-buffered GRU-output line with zero halos; [3] inner stride is
  // coprime with 64 banks -> conflict-free neighbor reads.
  __shared__ float hbuf[2][WW + 2][3];
  if (t < 2) {
    const int e = t * (WW + 1);      // 0 or WW+1
#pragma unroll
    for (int bf = 0; bf < 2; ++bf)
#pragma unroll
      for (int c = 0; c < 3; ++c) hbuf[bf][e][c] = 0.0f;
  }

  const float* xb = x   + (size_t)b * (3 * HWD);
  float*       cx = ctx + (size_t)b * (3 * HWD);

  const int sstep = dir ? 1 : WW;    // element step per scan iteration
  const int base  = dir ? t * WW : t;

  float h0 = 0.f, h1 = 0.f, h2 = 0.f;
  float xc0 = xb[base], xc1 = xb[HWD + base], xc2 = xb[2 * HWD + base];

  for (int s = 0; s < 512; ++s) {
    const int idx = base + s * sstep;

    // software prefetch of next step's input (hidden under this step's math)
    float xn0 = 0.f, xn1 = 0.f, xn2 = 0.f;
    if (s + 1 < 512) {
      const int idn = idx + sstep;
      xn0 = xb[idn]; xn1 = xb[HWD + idn]; xn2 = xb[2 * HWD + idn];
    }

    // GRU gates (PyTorch order r,z,n)
    float gi[9], gh[9];
#pragma unroll
    for (int j = 0; j < 9; ++j) {
      gi[j] = fmaf(Wi[j][0], xc0, fmaf(Wi[j][1], xc1, fmaf(Wi[j][2], xc2, Bi[j])));
      gh[j] = fmaf(Wh[j][0], h0,  fmaf(Wh[j][1], h1,  fmaf(Wh[j][2], h2,  Bh[j])));
    }
    const float hp[3] = {h0, h1, h2};
    float hn[3];
#pragma unroll
    for (int c = 0; c < 3; ++c) {
      const float r = fsig(gi[c] + gh[c]);
      const float z = fsig(gi[3 + c] + gh[3 + c]);
      const float n = ftanh_fast(fmaf(r, gh[6 + c], gi[6 + c]));
      hn[c] = fmaf(z, hp[c] - n, n);           // (1-z)*n + z*h
    }

    // ctx output = pre-conv hidden (coalesced for rows; L2-absorbed for cols)
    cx[idx]           = hn[0];
    cx[HWD + idx]     = hn[1];
    cx[2 * HWD + idx] = hn[2];

    const int bf = s & 1;
    hbuf[bf][t + 1][0] = hn[0];
    hbuf[bf][t + 1][1] = hn[1];
    hbuf[bf][t + 1][2] = hn[2];
    __syncthreads();                           // single barrier per step

    // width-3 conv over the line + tanh -> next hidden state
#pragma unroll
    for (int o = 0; o < 3; ++o) {
      float acc = Cb[o];
#pragma unroll
      for (int k = 0; k < 3; ++k)
#pragma unroll
        for (int ci = 0; ci < 3; ++ci)
          acc = fmaf(Cw[o][ci][k], hbuf[bf][t + k][ci], acc);
      const float hv = ftanh_fast(acc);
      if (o == 0) h0 = hv; else if (o == 1) h1 = hv; else h2 = hv;
    }

    xc0 = xn0; xc1 = xn1; xc2 = xn2;
  }
}

// ---------------------------------------------------------------------------
// Combine: out = sigmoid(W[3x6] . [ctx_above; ctx_left] + b), in place on
// d_out (which holds ctx_left). Bandwidth-bound, so expressing the 6-deep
// contraction as two chained V_WMMA_F32_16X16X4_F32 is free and exercises the
// CDNA5 matrix path. Each wave owns 16 pixels:
//   B (4x16): lanes 0-15 hold K0,K1; lanes 16-31 hold K2,K3 (per ISA layout)
//   A (16x4): lane m<3 carries a weight row, rows 3-15 zero-padded
//   C/D     : lane n, vgpr r -> (M=r+16*half/2, N=n); rows 0-2 used
// ---------------------------------------------------------------------------
#if defined(__has_builtin)
#if __has_builtin(__builtin_amdgcn_wmma_f32_16x16x4_f32)
#define CSRN_WMMA 1
#endif
#endif

__global__ __launch_bounds__(256) void csrn_combine(
    const float* __restrict__ ctx_above,
    const float* __restrict__ cwgt,    // (3,6)
    const float* __restrict__ cbias,   // (3)
    float* __restrict__ out)           // in: ctx_left; out: final
{
  const int lane = threadIdx.x & 31;
  const int n    = lane & 15;
  const int half = lane >> 4;
  const long grp = ((long)blockIdx.x * blockDim.x + threadIdx.x) >> 5;
  const long pix = grp * 16 + n;                // 16 pixels per wave
  const int  bb  = (int)(pix >> 18);            // / HWD
  const long off = pix & (long)(HWD - 1);
  const float* ca = ctx_above + (size_t)bb * (3 * HWD) + off;
  float*       cl = out       + (size_t)bb * (3 * HWD) + off;

#ifdef CSRN_WMMA
  v2f bm1, bm2, am1, am2;
  if (half == 0) {                 // K = 0,1 (chunk1)   K = 4,5 (chunk2)
    bm1[0] = ca[0];       bm1[1] = ca[HWD];
    bm2[0] = cl[HWD];     bm2[1] = cl[2 * HWD];
  } else {                         // K = 2,3 (chunk1)   K = 6,7 (chunk2, zero)
    bm1[0] = ca[2 * HWD]; bm1[1] = cl[0];
    bm2[0] = 0.f;         bm2[1] = 0.f;
  }
  am1[0] = 0.f; am1[1] = 0.f; am2[0] = 0.f; am2[1] = 0.f;
  if (n < 3) {                     // A row m = lane&15 for both halves
    const float* wr = cwgt + n * 6;
    if (half == 0) { am1[0] = wr[0]; am1[1] = wr[1]; am2[0] = wr[4]; am2[1] = wr[5]; }
    else           { am1[0] = wr[2]; am1[1] = wr[3]; }
  }
  v8f acc;
#pragma unroll
  for (int i = 0; i < 8; ++i) acc[i] = 0.f;
  if (half == 0) { acc[0] = cbias[0]; acc[1] = cbias[1]; acc[2] = cbias[2]; }

  v8f d = __builtin_amdgcn_wmma_f32_16x16x4_f32(false, am1, false, bm1,
                                                (short)0, acc, false, false);
  d     = __builtin_amdgcn_wmma_f32_16x16x4_f32(false, am2, false, bm2,
                                                (short)0, d,   false, false);
  if (half == 0) {
    cl[0]       = fsig(d[0]);
    cl[HWD]     = fsig(d[1]);
    cl[2 * HWD] = fsig(d[2]);
  }
#else
  if (half == 0) {
    float v[6];
    v[0] = ca[0]; v[1] = ca[HWD]; v[2] = ca[2 * HWD];
    v[3] = cl[0]; v[4] = cl[HWD]; v[5] = cl[2 * HWD];
    float o[3];
#pragma unroll
    for (int oo = 0; oo < 3; ++oo) {
      float a = cbias[oo];
#pragma unroll
      for (int c = 0; c < 6; ++c) a = fmaf(cwgt[oo * 6 + c], v[c], a);
      o[oo] = fsig(a);
    }
    cl[0] = o[0]; cl[HWD] = o[1]; cl[2 * HWD] = o[2];
  }
#endif
}

// ---------------------------------------------------------------------------
extern "C" void kernel_launch(void* const* d_in, const int* in_sizes, int n_in,
                              void* d_out, int out_size, void* d_ws, size_t ws_size,
                              hipStream_t stream)
{
  (void)in_sizes; (void)n_in; (void)out_size; (void)ws_size;
  const float* x     = (const float*)d_in[0];
  const float* wih_r = (const float*)d_in[1];
  const float* whh_r = (const float*)d_in[2];
  const float* bih_r = (const float*)d_in[3];
  const float* bhh_r = (const float*)d_in[4];
  const float* cnw_r = (const float*)d_in[5];
  const float* cnb_r = (const float*)d_in[6];
  const float* wih_c = (const float*)d_in[7];
  const float* whh_c = (const float*)d_in[8];
  const float* bih_c = (const float*)d_in[9];
  const float* bhh_c = (const float*)d_in[10];
  const float* cnw_c = (const float*)d_in[11];
  const float* cnb_c = (const float*)d_in[12];
  const float* cwgt  = (const float*)d_in[13];
  const float* cbias = (const float*)d_in[14];

  float* out       = (float*)d_out;
  float* ctx_above = (float*)d_ws;   // NB*3*HWD floats of scratch (== out_size)

  dim3 grid_scan(NB, 2);             // (batch, direction): 64 workgroups
  csrn_scan<<<grid_scan, 512, 0, stream>>>(x,
      wih_r, whh_r, bih_r, bhh_r, cnw_r, cnb_r,
      wih_c, whh_c, bih_c, bhh_c, cnw_c, cnb_c,
      ctx_above, out);

  const int blocks = (NB * HWD) / 128;   // 8 waves/block * 16 pixels/wave
  csrn_combine<<<blocks, 256, 0, stream>>>(ctx_above, cwgt, cbias, out);
}